// LinearSoftmaxAttention_74354473828785
// MI455X (gfx1250) — compile-verified
//
#include <hip/hip_runtime.h>
#include <math.h>

// Problem constants from the reference: N=1, L=S=512, H=8, D=M=32
#define H 8
#define L 512
#define S 512
#define D 32
#define LN_EPS 1e-5f

typedef __attribute__((ext_vector_type(2))) float v2f;
typedef __attribute__((ext_vector_type(8))) float v8f;

__device__ __forceinline__ float wsum(float v) {
  // wave32 all-lane sum
  #pragma unroll
  for (int m = 16; m > 0; m >>= 1) v += __shfl_xor(v, m, 32);
  return v;
}

// ---------------------------------------------------------------------------
// Kernel 1: LayerNorm(Q), LayerNorm(K)*scale*key_len, and head-major repack.
// One wave per row of 32 channels; lane = channel index.
// Input layout  : [l][h][d]  (rid = l*H + h)
// Output layout : [h][l][d]  (head-major for clean per-head WMMA tiles)
// ---------------------------------------------------------------------------
__global__ void k_normalize(const float* __restrict__ q,
                            const float* __restrict__ k,
                            const float* __restrict__ v,
                            const float* __restrict__ klen,
                            float* __restrict__ Qn,
                            float* __restrict__ Kn,
                            float* __restrict__ Vt) {
  const int lane = threadIdx.x;                                // d
  const int rid  = blockIdx.x * blockDim.y + threadIdx.y;      // l*H + h
  const int h = rid & (H - 1);
  const int l = rid >> 3;
  const int in_idx  = rid * D + lane;
  const int out_idx = (h * L + l) * D + lane;

  // queries -> layernorm
  float x  = q[in_idx];
  float mu = wsum(x) * (1.0f / D);
  float xm = x - mu;
  float var = wsum(xm * xm) * (1.0f / D);
  Qn[out_idx] = xm / sqrtf(var + LN_EPS);

  // keys -> layernorm, scaled by 1/(ALPHA*sqrt(D)) and key_lengths
  float y  = k[in_idx];
  float mk = wsum(y) * (1.0f / D);
  float ym = y - mk;
  float vk = wsum(ym * ym) * (1.0f / D);
  float kn = (ym / sqrtf(vk + LN_EPS)) * (1.0f / (3.0f * sqrtf(32.0f))) * klen[l];
  Kn[out_idx] = kn;

  // values -> head-major copy
  Vt[out_idx] = v[in_idx];
}

// ---------------------------------------------------------------------------
// WMMA fragment loader for C = X^T * Y style products where X,Y are [S][D]
// (row-major, D=32). Produces the A/B 16x4 f32 fragment for rows s0..s0+3,
// cols off..off+15. ISA layout (16x4 f32): lanes 0-15 -> K=0 (v0), K=1 (v1);
// lanes 16-31 -> K=2 (v0), K=3 (v1); M (or N) index = lane & 15.
// ---------------------------------------------------------------------------
__device__ __forceinline__ v2f load_frag_T(const float* __restrict__ base,
                                           int s0, int off, int lane) {
  const int hi = lane >> 4;
  const int c  = off + (lane & 15);
  const float* p = base + (s0 + 2 * hi) * D + c;
  v2f r;
  r.x = p[0];
  r.y = p[D];
  return r;
}

// ---------------------------------------------------------------------------
// Kernel 2: per-head 32x32 reductions with fp32 WMMA.
//   W1  = Kn^T * Vt   (used as order1 = Qn * W1)
//   Q2  = Qn^T * Qn
//   KKm = Kn^T * Kn
//   Ksum = sum_s Kn[s][:]
// One block per head, 4 waves = the four 16x16 output tiles.
// ---------------------------------------------------------------------------
__global__ void k_head_reduce(const float* __restrict__ Qn,
                              const float* __restrict__ Kn,
                              const float* __restrict__ Vt,
                              float* __restrict__ W1,
                              float* __restrict__ Q2,
                              float* __restrict__ KKm,
                              float* __restrict__ Ksum) {
  const int h    = blockIdx.x;
  const int wv   = threadIdx.x >> 5;
  const int lane = threadIdx.x & 31;
  const int ti   = (wv >> 1) * 16;   // output row offset (d index)
  const int tj   = (wv & 1) * 16;    // output col offset

  const float* Qh = Qn + h * (L * D);
  const float* Kh = Kn + h * (S * D);
  const float* Vh = Vt + h * (S * D);

  const float* Xs[3] = {Kh, Qh, Kh};
  const float* Ys[3] = {Vh, Qh, Kh};
  float*       Cs[3] = {W1 + h * D * D, Q2 + h * D * D, KKm + h * D * D};

  const int hi  = lane >> 4;
  const int col = tj + (lane & 15);

  #pragma unroll
  for (int p = 0; p < 3; ++p) {
    const float* X = Xs[p];
    const float* Y = Ys[p];
    v8f acc = {0.f, 0.f, 0.f, 0.f, 0.f, 0.f, 0.f, 0.f};
    for (int s0 = 0; s0 < S; s0 += 4) {
      v2f a = load_frag_T(X, s0, ti, lane);
      v2f b = load_frag_T(Y, s0, tj, lane);
      acc = __builtin_amdgcn_wmma_f32_16x16x4_f32(
          /*neg_a=*/false, a, /*neg_b=*/false, b,
          /*c_mod=*/(short)0, acc, /*reuse_a=*/false, /*reuse_b=*/false);
    }
    float* C = Cs[p];
    #pragma unroll
    for (int j = 0; j < 8; ++j) {
      C[(ti + j + 8 * hi) * D + col] = acc[j];  // C/D layout: row = j + 8*hi
    }
  }

  // Ksum by wave 0 (no WMMA under divergence)
  if (threadIdx.x < 32) {
    float s = 0.f;
    for (int ss = 0; ss < S; ++ss) s += Kh[ss * D + lane];
    Ksum[h * D + lane] = s;
  }
}

// ---------------------------------------------------------------------------
// Kernel 3: per-row scalars.
//   norm[h,l] = Q_l . Ksum + 0.5 * Q_l^T * KK * Q_l
//   c2[h,l]   = K_l^T * Q2 * K_l
// One wave per row; lane = channel e; shfl-broadcast for the d loop.
// ---------------------------------------------------------------------------
__global__ void k_row_scalars(const float* __restrict__ Qn,
                              const float* __restrict__ Kn,
                              const float* __restrict__ Q2,
                              const float* __restrict__ KKm,
                              const float* __restrict__ Ksum,
                              float* __restrict__ normArr,
                              float* __restrict__ c2Arr) {
  const int lane = threadIdx.x;                                // e
  const int rid  = blockIdx.x * blockDim.y + threadIdx.y;      // h*L + l
  const int h = rid >> 9;                                      // L = 512

  const float qv = Qn[rid * D + lane];
  const float kv = Kn[rid * D + lane];
  const float* KKh = KKm + h * D * D;
  const float* Q2h = Q2  + h * D * D;

  float t1 = 0.f, t2 = 0.f;
  #pragma unroll
  for (int d = 0; d < D; ++d) {
    float qd = __shfl(qv, d, 32);
    float kd = __shfl(kv, d, 32);
    t1 = fmaf(qd, KKh[d * D + lane], t1);
    t2 = fmaf(kd, Q2h[d * D + lane], t2);
  }
  float npart = qv * Ksum[h * D + lane] + 0.5f * qv * t1;
  float cpart = kv * t2;
  npart = wsum(npart);
  cpart = wsum(cpart);
  if (lane == 0) {
    normArr[rid] = npart;
    c2Arr[rid]   = cpart;
  }
}

// ---------------------------------------------------------------------------
// Kernel 4: order1 = Qn * W1 via WMMA (K = 32 -> 8 steps), fused epilogue:
//   out[l,h,m] = (order1[l,m] + 0.5*c2[l]*V[l,h,m]) / norm[l]
// Grid: (L/16, H); 2 waves per block, wave wv owns the 16-col slab m0=16*wv.
// ---------------------------------------------------------------------------
__global__ void k_output(const float* __restrict__ Qn,
                         const float* __restrict__ W1,
                         const float* __restrict__ vals,   // original values [l][h][m]
                         const float* __restrict__ normArr,
                         const float* __restrict__ c2Arr,
                         float* __restrict__ out) {
  const int h    = blockIdx.y;
  const int l0   = blockIdx.x * 16;
  const int wv   = threadIdx.x >> 5;
  const int lane = threadIdx.x & 31;
  const int m0   = wv * 16;
  const int hi   = lane >> 4;
  const int lrow = lane & 15;

  const float* Qh = Qn + h * (L * D);   // [512][32] row-major
  const float* Wh = W1 + h * (D * D);   // [32][32]  d-major

  v8f acc = {0.f, 0.f, 0.f, 0.f, 0.f, 0.f, 0.f, 0.f};
  #pragma unroll
  for (int d0 = 0; d0 < D; d0 += 4) {
    // A fragment: rows l0..l0+15 of Q, K-chunk d0..d0+3 (contiguous in memory)
    const float* pa = Qh + (l0 + lrow) * D + d0 + 2 * hi;
    v2f a; a.x = pa[0]; a.y = pa[1];
    // B fragment: rows d0..d0+3 of W1, cols m0..m0+15
    const float* pb = Wh + (d0 + 2 * hi) * D + m0 + lrow;
    v2f b; b.x = pb[0]; b.y = pb[D];
    acc = __builtin_amdgcn_wmma_f32_16x16x4_f32(
        false, a, false, b, (short)0, acc, false, false);
  }

  #pragma unroll
  for (int j = 0; j < 8; ++j) {
    const int r = l0 + j + 8 * hi;             // output row l
    const int m = m0 + lrow;                   // output col
    const int gidx = (r * H + h) * D + m;      // [l][h][m] flat
    const float nrm = normArr[h * L + r];
    const float c2  = c2Arr[h * L + r];
    out[gidx] = (acc[j] + 0.5f * c2 * vals[gidx]) / nrm;
  }
}

// ---------------------------------------------------------------------------
// Host launcher
// ---------------------------------------------------------------------------
extern "C" void kernel_launch(void* const* d_in, const int* in_sizes, int n_in,
                              void* d_out, int out_size, void* d_ws, size_t ws_size,
                              hipStream_t stream) {
  const float* q    = (const float*)d_in[0];  // queries     [1,512,8,32]
  const float* k    = (const float*)d_in[1];  // keys        [1,512,8,32]
  const float* v    = (const float*)d_in[2];  // values      [1,512,8,32]
  // d_in[3] attn_mask (unused by reference), d_in[4] query_lengths (unused)
  const float* klen = (const float*)d_in[5];  // key_lengths [1,512]
  float* out = (float*)d_out;

  float* ws = (float*)d_ws;
  float* Qn      = ws;                 // H*L*D = 131072
  float* Kn      = Qn + H * L * D;     // 131072
  float* Vt      = Kn + H * S * D;     // 131072
  float* W1      = Vt + H * S * D;     // H*D*D = 8192
  float* Q2      = W1 + H * D * D;     // 8192
  float* KKm     = Q2 + H * D * D;     // 8192
  float* Ksum    = KKm + H * D * D;    // H*D = 256
  float* normArr = Ksum + H * D;       // H*L = 4096
  float* c2Arr   = normArr + H * L;    // 4096   (total ~1.7 MB)

  dim3 rowBlk(32, 8);  // one wave32 per row, 8 rows per block
  k_normalize  <<<(H * L) / 8, rowBlk, 0, stream>>>(q, k, v, klen, Qn, Kn, Vt);
  k_head_reduce<<<H, 128, 0, stream>>>(Qn, Kn, Vt, W1, Q2, KKm, Ksum);
  k_row_scalars<<<(H * L) / 8, rowBlk, 0, stream>>>(Qn, Kn, Q2, KKm, Ksum, normArr, c2Arr);
  k_output     <<<dim3(L / 16, H), 64, 0, stream>>>(Qn, W1, v, normArr, c2Arr, out);
}